// STDPConv2dSimple_78632261255759
// MI455X (gfx1250) — compile-verified
//
#include <hip/hip_runtime.h>

typedef float v2f __attribute__((ext_vector_type(2)));
typedef float v8f __attribute__((ext_vector_type(8)));

#define B_N      32
#define C_IN     64
#define C_OUT    128
#define HW       64
#define KVOL     576            // 64*3*3
#define SPB      524288         // C_OUT*HW*HW per batch
#define THRESH   15.0f
#define SW_PITCH 580            // 576 padded: even (b64 align), 580%64=4 spread
#define CHUNK_CI 8
#define KCH      (CHUNK_CI * 9) // 72 K values per ci-chunk
#define SB_PITCH 74             // 72 padded: even (b64 align), 74%64=10 spread

// ---------------------------------------------------------------------------
// Kernel 1: conv2d via implicit GEMM with V_WMMA_F32_16X16X4_F32.
// grid = (B*H, C_OUT/16); block = 64 threads = 2 waves.
// Each wave computes a 16(Cout) x 32(w) tile as two 16x16 WMMA tiles that
// share the A (weight) fragment: per K-step = 1 A ds_load_b64 +
// 2 B ds_load_b64 + 2 v_wmma, all with compile-time immediate offsets.
// Input is staged in im2col layout sB[n][k]; next chunk prefetched.
// ---------------------------------------------------------------------------
__global__ __launch_bounds__(64) void conv_wmma_f32(
    const float* __restrict__ x, const float* __restrict__ w,
    float* __restrict__ out) {
  __shared__ float sW[16 * SW_PITCH];   // 16 couts x 576 (padded)
  __shared__ float sB[HW * SB_PITCH];   // 64 w-positions x 72 K (padded)

  const int t   = threadIdx.x;
  const int bh  = blockIdx.x;           // b*64 + h
  const int b   = bh >> 6;
  const int h   = bh & 63;
  const int co0 = blockIdx.y << 4;

  // ---- Stage 16x576 weight slice: b128 loads -> b128 LDS stores, no divides.
  {
    const int m  = t >> 2;              // 16 rows, 4 threads per row
    const int kq = (t & 3) << 2;        // starting col: 0,4,8,12
    const float* wrow = w + (size_t)(co0 + m) * KVOL;
#pragma unroll
    for (int j = 0; j < 36; ++j) {
      int k = kq + j * 16;
      *(float4*)&sW[m * SW_PITCH + k] = *(const float4*)(wrow + k);
    }
  }

  const int lane = t & 31;
  const int wave = t >> 5;              // 0..1
  const int half = lane >> 4;           // 0: K {k,k+1}, 1: K {k+2,k+3}
  const int lm   = lane & 15;

  const int aBase  = lm * SW_PITCH + (half << 1);
  const int bBase0 = ((wave << 5) + lm) * SB_PITCH + (half << 1);
  const int bBase1 = bBase0 + 16 * SB_PITCH;

  v8f c0 = {}, c1 = {};

  const float* xb0 = x + (size_t)b * C_IN * (HW * HW);

  for (int cblk = 0; cblk < C_IN / CHUNK_CI; ++cblk) {
    const float* xb = xb0 + (size_t)cblk * CHUNK_CI * (HW * HW);
    __syncthreads();
    // ---- im2col staging: one thread per w-position, fully unrolled
    // (ci,kh,kw) so LDS offsets are immediates and bounds checks CSE.
    {
      const int nn = t;                 // 0..63
#pragma unroll
      for (int ci = 0; ci < CHUNK_CI; ++ci) {
#pragma unroll
        for (int kh = 0; kh < 3; ++kh) {
          const int row = h + kh - 1;
          const bool rok = (unsigned)row < (unsigned)HW;
#pragma unroll
          for (int kw = 0; kw < 3; ++kw) {
            const int wx = nn + kw - 1;
            float v = 0.0f;
            if (rok && (unsigned)wx < (unsigned)HW)
              v = xb[ci * (HW * HW) + row * HW + wx];
            sB[nn * SB_PITCH + ci * 9 + kh * 3 + kw] = v;
          }
        }
      }
    }
    __syncthreads();

    // ---- Prefetch next chunk's input rows into cache (global_prefetch_b8).
    if (cblk < C_IN / CHUNK_CI - 1 && t < 24) {
      int ci = t / 3;
      int kh = t - ci * 3;
      int row = h + kh - 1;
      row = row < 0 ? 0 : (row > HW - 1 ? HW - 1 : row);
      __builtin_prefetch(xb + (CHUNK_CI + ci) * (HW * HW) + row * HW, 0, 0);
    }

    // ---- Hot loop: 18 steps x (1 A b64 + 2 B b64 + 2 wmma).
    const float* sWk = sW + aBase + cblk * KCH;
    const float* sB0 = sB + bBase0;
    const float* sB1 = sB + bBase1;
#pragma unroll
    for (int k0 = 0; k0 < KCH; k0 += 4) {
      v2f a  = *(const v2f*)(sWk + k0);
      v2f b0 = *(const v2f*)(sB0 + k0);
      v2f b1 = *(const v2f*)(sB1 + k0);
      c0 = __builtin_amdgcn_wmma_f32_16x16x4_f32(
          false, a, false, b0, (short)0, c0, false, false);
      c1 = __builtin_amdgcn_wmma_f32_16x16x4_f32(
          false, a, false, b1, (short)0, c1, false, false);
    }
  }

  // C/D layout: VGPR r -> M=r (lanes 0-15) / M=8+r (lanes 16-31), N=lane&15.
  const size_t outBase =
      (((size_t)b * C_OUT + co0 + (half << 3)) * HW + h) * HW + (wave << 5) + lm;
#pragma unroll
  for (int r = 0; r < 8; ++r) {
    out[outBase + (size_t)r * (HW * HW)]      = c0[r];
    out[outBase + (size_t)r * (HW * HW) + 16] = c1[r];
  }
}

// ---------------------------------------------------------------------------
// Kernel 2: per-block partial top-5. grid = B*64; each block scans 8192
// contiguous elements of one batch with coalesced strided reads.
// ---------------------------------------------------------------------------
__device__ __forceinline__ void ins5(float v, int idx, float* tv, int* ti) {
  if (v > tv[4]) {
    int p = 4;
    while (p > 0 && v > tv[p - 1]) {
      tv[p] = tv[p - 1]; ti[p] = ti[p - 1]; --p;
    }
    tv[p] = v; ti[p] = idx;
  }
}

__global__ __launch_bounds__(256) void topk_partial(
    const float* __restrict__ pot, float* __restrict__ pVal,
    int* __restrict__ pIdx) {
  __shared__ float sv[256 * 5];
  __shared__ int   si[256 * 5];
  const int t     = threadIdx.x;
  const int blk   = blockIdx.x;       // b*64 + chunk
  const int b     = blk >> 6;
  const int chunk = blk & 63;
  const int base  = chunk * 8192;
  const float* p  = pot + (size_t)b * SPB;

  float tv[5]; int ti[5];
#pragma unroll
  for (int j = 0; j < 5; ++j) { tv[j] = -1e30f; ti[j] = -1; }

  for (int i = 0; i < 32; ++i) {
    int idx = base + t + i * 256;
    ins5(p[idx], idx, tv, ti);
  }
#pragma unroll
  for (int j = 0; j < 5; ++j) { sv[t * 5 + j] = tv[j]; si[t * 5 + j] = ti[j]; }
  __syncthreads();

  if (t == 0) {
    float fv[5]; int fi[5];
    for (int j = 0; j < 5; ++j) { fv[j] = -1e30f; fi[j] = -1; }
    for (int e = 0; e < 256 * 5; ++e) ins5(sv[e], si[e], fv, fi);
    for (int j = 0; j < 5; ++j) {
      pVal[blk * 5 + j] = fv[j];
      pIdx[blk * 5 + j] = fi[j];
    }
  }
}

// Kernel 3: final merge, one thread per batch over 64*5 = 320 candidates.
__global__ void topk_final(const float* __restrict__ pVal,
                           const int* __restrict__ pIdx,
                           float* __restrict__ fVal, int* __restrict__ fIdx) {
  const int b = threadIdx.x;          // 32 threads
  float fv[5]; int fi[5];
  for (int j = 0; j < 5; ++j) { fv[j] = -1e30f; fi[j] = -1; }
  for (int e = 0; e < 320; ++e)
    ins5(pVal[b * 320 + e], pIdx[b * 320 + e], fv, fi);
  for (int j = 0; j < 5; ++j) {
    fVal[b * 5 + j] = fv[j];
    fIdx[b * 5 + j] = fi[j];
  }
}

// Kernel 4: zero output with 128-bit stores.
__global__ __launch_bounds__(256) void zero_out(float4* __restrict__ out) {
  out[(size_t)blockIdx.x * 256 + threadIdx.x] = make_float4(0.f, 0.f, 0.f, 0.f);
}

// Kernel 5: scatter fire=1.0 at winners above threshold.
__global__ void scatter_fire(const float* __restrict__ fVal,
                             const int* __restrict__ fIdx,
                             float* __restrict__ out) {
  const int t = threadIdx.x;          // 160 threads: 32 batches x 5 winners
  const int b = t / 5;
  float v = fVal[t];
  int idx = fIdx[t];
  if (v > THRESH && idx >= 0)
    out[(size_t)b * SPB + idx] = 1.0f;
}

extern "C" void kernel_launch(void* const* d_in, const int* in_sizes, int n_in,
                              void* d_out, int out_size, void* d_ws,
                              size_t ws_size, hipStream_t stream) {
  const float* x = (const float*)d_in[0];   // (32,64,64,64)
  const float* w = (const float*)d_in[1];   // (128,64,3,3)
  float* out = (float*)d_out;               // (32,128,64,64)

  // d_ws layout (floats): [0,10240) partVal | [10240,20480) partIdx
  //                       [20480,20640) finalVal | [20640,20800) finalIdx
  float* wsf  = (float*)d_ws;
  float* pVal = wsf;
  int*   pIdx = (int*)(wsf + 10240);
  float* fVal = wsf + 20480;
  int*   fIdx = (int*)(wsf + 20640);

  // 1. conv2d -> potentials in d_out
  conv_wmma_f32<<<dim3(B_N * HW, C_OUT / 16), dim3(64), 0, stream>>>(x, w, out);
  // 2. partial top-5 per 8192-element chunk
  topk_partial<<<dim3(B_N * 64), dim3(256), 0, stream>>>(out, pVal, pIdx);
  // 3. final per-batch top-5
  topk_final<<<dim3(1), dim3(32), 0, stream>>>(pVal, pIdx, fVal, fIdx);
  // 4. zero the output (16777216 floats = 4194304 float4)
  zero_out<<<dim3(16384), dim3(256), 0, stream>>>((float4*)out);
  // 5. scatter winners
  scatter_fire<<<dim3(1), dim3(160), 0, stream>>>(fVal, fIdx, out);
}